// KeyFormer_31731218382920
// MI455X (gfx1250) — compile-verified
//
#include <hip/hip_runtime.h>
#include <math.h>

// ---------------------------------------------------------------------------
// Problem constants (match reference)
// ---------------------------------------------------------------------------
#define M_DIM 512
#define N_DIM 4096
#define D_DIM 128
#define H_DIM 32
#define L_DIM 4096
#define P_OFF 3584

// GEMM tiling: block tile 128x64, 8 wave32 in 4x2 grid, 32x32 C per wave
#define BM 128
#define BN 64
#define BK 32
#define LDA_S  36    // padded LDS stride (floats) for A tile rows
#define LDBI_S 132   // interleaved B tile: (2*BN + 4) floats per k-pair row

typedef __attribute__((ext_vector_type(2))) float v2f;
typedef __attribute__((ext_vector_type(8))) float v8f;

__device__ __forceinline__ v8f wmma4(v2f a, v2f b, v8f c) {
  // V_WMMA_F32_16X16X4_F32 : D = A(16x4,f32) * B(4x16,f32) + C(16x16,f32)
  return __builtin_amdgcn_wmma_f32_16x16x4_f32(
      /*neg_a=*/false, a, /*neg_b=*/false, b,
      /*c_mod=*/(short)0, c, /*reuse_a=*/false, /*reuse_b=*/false);
}

// ---------------------------------------------------------------------------
// Generic WMMA GEMM:  C(BMxBN tile) = A(MxK) * B(KxN)
//   BMODE 0: B row-major [K][N], ldb = row stride
//   BMODE 1: B is transposed source: Bmat[k][n] = Bsrc[n*ldb + k]  (q·K^T)
//   SMODE 0: qkv store   -> out[h*headStride + (rowOff+m)*D + d], (h,d)=n/128,n%128
//   SMODE 1: scores store-> out[head*M*L + m*L + n]
//   SMODE 2: output store-> out[m*N + head*D + n]
// B tile is staged k-pair interleaved: Bs[(k>>1)*LDBI_S + 2*n + (k&1)] so a
// fragment (B[k][n], B[k+1][n]) is one aligned 64-bit LDS load.
// blockDim = 256 (8 wave32), grid = (N/BN, M/BM, heads)
// ---------------------------------------------------------------------------
template <int BMODE, int SMODE>
__global__ __launch_bounds__(256)
void gemm_wmma(const float* __restrict__ A, const float* __restrict__ B,
               float* __restrict__ C, int Kdim, int lda, int ldb,
               int headStride, int rowOff)
{
  __shared__ float As[BM * LDA_S];
  __shared__ float Bs[(BK / 2) * LDBI_S];

  const int tid  = threadIdx.x;
  const int lane = tid & 31;
  const int wave = tid >> 5;
  const int head = blockIdx.z;

  const int nBase = blockIdx.x * BN;
  const int mBase = blockIdx.y * BM;

  if (SMODE == 1) {  // scores: per-head q and K
    A += (size_t)head * M_DIM * D_DIM;
    B += (size_t)head * L_DIM * D_DIM;
  }
  if (SMODE == 2) {  // output: per-head weights and V
    A += (size_t)head * M_DIM * L_DIM;
    B += (size_t)head * L_DIM * D_DIM;
  }

  const int waveM = wave >> 1;        // 0..3 -> 32-row strip
  const int waveN = wave & 1;         // 0..1 -> 32-col strip
  const int l15   = lane & 15;
  const int kb    = (lane >> 4) * 2;  // lanes 16-31 hold K+2,K+3
  const int hi16  = 8 * (lane >> 4);

  v8f acc[2][2] = {};

  for (int kBase = 0; kBase < Kdim; kBase += BK) {
    // ---- stage A tile: BM x BK (row-major, float4 loads) ----
    #pragma unroll
    for (int i = tid; i < (BM * BK) / 4; i += 256) {   // 1024 -> 4 per thread
      const int row = i >> 3;        // 8 float4 per row
      const int c4  = i & 7;
      float4 v = *(const float4*)(A + (size_t)(mBase + row) * lda + kBase + c4 * 4);
      *(float4*)(&As[row * LDA_S + c4 * 4]) = v;
    }
    // ---- stage B tile: BK x BN, k-pair interleaved ----
    if (BMODE == 0) {
      #pragma unroll
      for (int i = tid; i < (BK * BN) / 4; i += 256) { // 512 -> 2 per thread
        const int row = i >> 4;      // k, 16 float4 per row
        const int c4  = i & 15;
        float4 v = *(const float4*)(B + (size_t)(kBase + row) * ldb + nBase + c4 * 4);
        float* dst = &Bs[(row >> 1) * LDBI_S + 2 * (c4 * 4) + (row & 1)];
        dst[0] = v.x; dst[2] = v.y; dst[4] = v.z; dst[6] = v.w;
      }
    } else {
      // transposed source: memory row (fixed n = l) is contiguous in k = d
      #pragma unroll
      for (int i = tid; i < (BN * BK) / 4; i += 256) { // 512 -> 2 per thread
        const int n  = i >> 3;       // 8 float4 per n-row
        const int k4 = i & 7;        // k = 4*k4 .. 4*k4+3
        float4 v = *(const float4*)(B + (size_t)(nBase + n) * ldb + kBase + k4 * 4);
        float2 lo; lo.x = v.x; lo.y = v.y;   // k even/odd pair (4k4, 4k4+1)
        float2 hi; hi.x = v.z; hi.y = v.w;   // pair (4k4+2, 4k4+3)
        *(float2*)(&Bs[(2 * k4 + 0) * LDBI_S + 2 * n]) = lo;
        *(float2*)(&Bs[(2 * k4 + 1) * LDBI_S + 2 * n]) = hi;
      }
    }
    // prefetch next K tile while computing (global_prefetch_b8)
    if (kBase + BK < Kdim) {
      __builtin_prefetch(A + (size_t)(mBase + (tid & 127)) * lda + kBase + BK, 0, 0);
      if (BMODE == 0)
        __builtin_prefetch(B + (size_t)(kBase + BK + (tid & 31)) * ldb + nBase, 0, 0);
      else
        __builtin_prefetch(B + (size_t)(nBase + (tid & 63)) * ldb + kBase + BK, 0, 0);
    }
    __syncthreads();

    // ---- compute: 8 k-steps, 4 WMMA per step per wave (32x32 C tile) ----
    #pragma unroll
    for (int kk = 0; kk < BK; kk += 4) {
      const int krow = (kk + kb) >> 1;
      v2f a0 = *(const v2f*)(&As[(waveM * 32 +      l15) * LDA_S + kk + kb]);
      v2f a1 = *(const v2f*)(&As[(waveM * 32 + 16 + l15) * LDA_S + kk + kb]);
      v2f b0 = *(const v2f*)(&Bs[krow * LDBI_S + 2 * (waveN * 32 +      l15)]);
      v2f b1 = *(const v2f*)(&Bs[krow * LDBI_S + 2 * (waveN * 32 + 16 + l15)]);
      acc[0][0] = wmma4(a0, b0, acc[0][0]);
      acc[0][1] = wmma4(a0, b1, acc[0][1]);
      acc[1][0] = wmma4(a1, b0, acc[1][0]);
      acc[1][1] = wmma4(a1, b1, acc[1][1]);
    }
    __syncthreads();
  }

  // ---- store C: VGPR r holds rows +r (lanes 0-15) / +r+8 (lanes 16-31) ----
  #pragma unroll
  for (int mt = 0; mt < 2; ++mt) {
    #pragma unroll
    for (int nt = 0; nt < 2; ++nt) {
      #pragma unroll
      for (int r = 0; r < 8; ++r) {
        const int m  = mBase + waveM * 32 + mt * 16 + r + hi16;
        const int nn = nBase + waveN * 32 + nt * 16 + l15;
        const float v = acc[mt][nt][r];
        if (SMODE == 0) {
          const int h = nn >> 7, d = nn & 127;
          C[(size_t)h * headStride + (size_t)(rowOff + m) * D_DIM + d] = v;
        } else if (SMODE == 1) {
          float* Ch = C + (size_t)head * M_DIM * L_DIM;
          Ch[(size_t)m * L_DIM + nn] = v;
        } else {
          C[(size_t)m * N_DIM + head * D_DIM + nn] = v;
        }
      }
    }
  }
}

// ---------------------------------------------------------------------------
// Dual softmax over rows of length L: weights (in-place into scores) and
// perturb_out = softmax((scores+noise)/1.5). One block per (h,m) row.
// ---------------------------------------------------------------------------
__global__ __launch_bounds__(256)
void softmax_dual(float* __restrict__ scores, const float* __restrict__ noise,
                  float* __restrict__ perturb_out)
{
  const int    tid  = threadIdx.x;
  const size_t base = (size_t)blockIdx.x * L_DIM;

  float e1[16], e2[16];
  float s1 = 0.f, s2 = 0.f;
  #pragma unroll
  for (int i = 0; i < 16; ++i) {
    const int idx = tid + i * 256;
    const float s = scores[base + idx];
    const float p = (s + noise[base + idx]) * (1.0f / 1.5f);
    e1[i] = __expf(s);
    e2[i] = __expf(p);
    s1 += e1[i];
    s2 += e2[i];
  }

  __shared__ float r1[256];
  __shared__ float r2[256];
  r1[tid] = s1;
  r2[tid] = s2;
  __syncthreads();
  #pragma unroll
  for (int off = 128; off > 0; off >>= 1) {
    if (tid < off) {
      r1[tid] += r1[tid + off];
      r2[tid] += r2[tid + off];
    }
    __syncthreads();
  }
  const float inv1 = 1.0f / r1[0];
  const float inv2 = 1.0f / r2[0];

  #pragma unroll
  for (int i = 0; i < 16; ++i) {
    const int idx = tid + i * 256;
    scores[base + idx]      = e1[i] * inv1;   // weights, reused by weights·V GEMM
    perturb_out[base + idx] = e2[i] * inv2;
  }
}

// ---------------------------------------------------------------------------
// Host launcher
// ---------------------------------------------------------------------------
extern "C" void kernel_launch(void* const* d_in, const int* in_sizes, int n_in,
                              void* d_out, int out_size, void* d_ws, size_t ws_size,
                              hipStream_t stream)
{
  (void)in_sizes; (void)n_in; (void)out_size; (void)ws_size;

  const float* X     = (const float*)d_in[0];
  const float* W_q   = (const float*)d_in[1];
  const float* W_k   = (const float*)d_in[2];
  const float* W_v   = (const float*)d_in[3];
  const float* noise = (const float*)d_in[4];
  const float* cK    = (const float*)d_in[5];
  const float* cV    = (const float*)d_in[6];
  // d_in[7] = P (static 3584, baked in as P_OFF)

  // workspace layout (floats)
  float* ws = (float*)d_ws;
  float* q  = ws;                                        //  H*M*D  (8 MB)
  float* Kc = q  + (size_t)H_DIM * M_DIM * D_DIM;        //  H*L*D  (64 MB)
  float* Vc = Kc + (size_t)H_DIM * L_DIM * D_DIM;        //  H*L*D  (64 MB)
  float* sc = Vc + (size_t)H_DIM * L_DIM * D_DIM;        //  H*M*L  (256 MB)

  float* out  = (float*)d_out;                           //  M*N
  float* pout = out + (size_t)M_DIM * N_DIM;             //  H*M*L

  const size_t cacheBytes = (size_t)H_DIM * L_DIM * D_DIM * sizeof(float);
  hipMemcpyAsync(Kc, cK, cacheBytes, hipMemcpyDeviceToDevice, stream);
  hipMemcpyAsync(Vc, cV, cacheBytes, hipMemcpyDeviceToDevice, stream);

  dim3 blk(256);

  // QKV projection: q -> q buffer, k/v -> appended rows [P, P+M) of cache copies
  dim3 g1(N_DIM / BN, M_DIM / BM, 1);
  gemm_wmma<0, 0><<<g1, blk, 0, stream>>>(X, W_q, q,  N_DIM, N_DIM, N_DIM, M_DIM * D_DIM, 0);
  gemm_wmma<0, 0><<<g1, blk, 0, stream>>>(X, W_k, Kc, N_DIM, N_DIM, N_DIM, L_DIM * D_DIM, P_OFF);
  gemm_wmma<0, 0><<<g1, blk, 0, stream>>>(X, W_v, Vc, N_DIM, N_DIM, N_DIM, L_DIM * D_DIM, P_OFF);

  // scores[h] = q[h] (512x128) · Kc[h]^T (128x4096)
  dim3 g2(L_DIM / BN, M_DIM / BM, H_DIM);
  gemm_wmma<1, 1><<<g2, blk, 0, stream>>>(q, Kc, sc, D_DIM, D_DIM, D_DIM, 0, 0);

  // dual softmax: weights in-place, perturb_out to d_out
  softmax_dual<<<dim3(H_DIM * M_DIM), blk, 0, stream>>>(sc, noise, pout);

  // output[h] = weights[h] (512x4096) · Vc[h] (4096x128)
  dim3 g4(D_DIM / BN, M_DIM / BM, H_DIM);
  gemm_wmma<0, 2><<<g4, blk, 0, stream>>>(sc, Vc, out, L_DIM, L_DIM, D_DIM, 0, 0);
}